// Autoregression_69355131895897
// MI455X (gfx1250) — compile-verified
//
#include <hip/hip_runtime.h>
#include <hip/hip_bf16.h>

typedef float v2f __attribute__((ext_vector_type(2)));
typedef float v8f __attribute__((ext_vector_type(8)));

#define NJ   23
#define MAXC 7

// [joint, ancestors...] padded with -1 (mask==0 slots in the reference).
__constant__ int c_idx[NJ][MAXC] = {
  { 0,-1,-1,-1,-1,-1,-1},
  { 1,-1,-1,-1,-1,-1,-1},
  { 2,-1,-1,-1,-1,-1,-1},
  { 3, 0,-1,-1,-1,-1,-1},
  { 4, 1,-1,-1,-1,-1,-1},
  { 5, 2,-1,-1,-1,-1,-1},
  { 6, 3, 0,-1,-1,-1,-1},
  { 7, 4, 1,-1,-1,-1,-1},
  { 8, 5, 2,-1,-1,-1,-1},
  { 9, 6, 3, 0,-1,-1,-1},
  {10, 7, 4, 1,-1,-1,-1},
  {11, 8, 5, 2,-1,-1,-1},
  {12, 8, 5, 2,-1,-1,-1},
  {13, 8, 5, 2,-1,-1,-1},
  {14,11, 8, 5, 2,-1,-1},
  {15,12, 8, 5, 2,-1,-1},
  {16,13, 8, 5, 2,-1,-1},
  {17,15,11, 8, 5, 2,-1},
  {18,16,12, 8, 5, 2,-1},
  {19,17,13, 8, 5, 2,-1},
  {20,18,15,11, 8, 5, 2},
  {21,19,16,12, 8, 5, 2},
  {22,20,17,13, 8, 5, 2}};

// One cyclic Jacobi rotation zeroing M[p][q]; accumulates eigenvectors into V columns.
__device__ __forceinline__ void jacobi_rot(float (&M)[3][3], float (&V)[3][3], int p, int q) {
  float apq = M[p][q];
  if (fabsf(apq) < 1e-20f) return;
  float tau = (M[q][q] - M[p][p]) / (2.f * apq);
  float t   = copysignf(1.f, tau) / (fabsf(tau) + sqrtf(1.f + tau * tau));
  float cc  = 1.f / sqrtf(1.f + t * t);
  float ss  = t * cc;
#pragma unroll
  for (int k = 0; k < 3; ++k) {        // M = M * G
    float mkp = M[k][p], mkq = M[k][q];
    M[k][p] = cc * mkp - ss * mkq;
    M[k][q] = ss * mkp + cc * mkq;
  }
#pragma unroll
  for (int k = 0; k < 3; ++k) {        // M = G^T * M
    float mpk = M[p][k], mqk = M[q][k];
    M[p][k] = cc * mpk - ss * mqk;
    M[q][k] = ss * mpk + cc * mqk;
  }
#pragma unroll
  for (int k = 0; k < 3; ++k) {        // V = V * G
    float vkp = V[k][p], vkq = V[k][q];
    V[k][p] = cc * vkp - ss * vkq;
    V[k][q] = ss * vkp + cc * vkq;
  }
}

__global__ __launch_bounds__(256)
void pose_fused_kernel(const float* __restrict__ feature,
                       const float* __restrict__ W1, const float* __restrict__ b1,
                       const float* __restrict__ W2, const float* __restrict__ b2,
                       const float* __restrict__ W3, const float* __restrict__ b3,
                       const float* __restrict__ Wp, const float* __restrict__ bp,
                       float* __restrict__ out) {
  __shared__ __align__(16) float xl[72];   // padded input (69 -> 72, zeros above 68)
  __shared__ __align__(16) float h1[128];
  __shared__ __align__(16) float h2[128];
  __shared__ __align__(16) float jF[72];   // joint_F, 69 used

  const int tid  = threadIdx.x;
  const int lane = tid & 31;
  const int wave = tid >> 5;
  const int m    = lane & 15;  // row/col index within 16
  const int half = lane >> 4;  // K-pair selector in f32 WMMA fragments

  // A-fragment mask: only batch row M==0 is real; rows 1..15 are zero padding.
  const float amaskf = (m == 0) ? 1.f : 0.f;
  const v2f   amask  = {amaskf, amaskf};

  // Warm L2/WGP$ for the later weights while layer 1 runs (global_prefetch_b8).
  for (int off = tid * 64; off < 128 * 128 * 4; off += 256 * 64)
    __builtin_prefetch((const char*)W2 + off, 0, 3);
  for (int off = tid * 64; off < 69 * 128 * 4; off += 256 * 64)
    __builtin_prefetch((const char*)W3 + off, 0, 3);

  for (int i = tid; i < 72; i += 256) xl[i] = (i < 69) ? feature[3 + i] : 0.f;
  __syncthreads();

  // ---- Layer 1: h1 = relu(x @ W1^T + b1). M=16 (row0 real), N=128, K=72 ----
  // K slots 69..71 have A==0 (zero-padded LDS), so B there is a don't-care:
  // clamp the load index to 68 to stay in-bounds instead of masking.
  {
    const int n0 = wave * 16;
    const float* Brow = W1 + (n0 + m) * 69;          // B[k][n] = W1[n][k]
    v2f bf[18], af[18];
#pragma unroll
    for (int s = 0; s < 18; ++s) {                   // issue all global loads first
      const int ka = 4 * s + 2 * half;
      const int i0 = (ka     <= 68) ? ka     : 68;
      const int i1 = (ka + 1 <= 68) ? ka + 1 : 68;
      bf[s][0] = Brow[i0];
      bf[s][1] = Brow[i1];
    }
#pragma unroll
    for (int s = 0; s < 18; ++s) {                   // then all LDS loads (ds_load_b64)
      const int ka = 4 * s + 2 * half;
      af[s] = *(const v2f*)(xl + ka) * amask;
    }
    v8f c = {};
#pragma unroll
    for (int s = 0; s < 18; ++s)                     // back-to-back WMMA accumulate
      c = __builtin_amdgcn_wmma_f32_16x16x4_f32(false, af[s], false, bf[s], (short)0, c, false, false);
    if (half == 0) {                                 // c[0] holds (M=0, N=lane)
      const int n = n0 + m;
      h1[n] = fmaxf(c[0] + b1[n], 0.f);
    }
  }
  __syncthreads();

  // ---- Layer 2: h2 = relu(h1 @ W2^T + b2). N=128, K=128 ----
  {
    const int n0 = wave * 16;
    const float* Brow = W2 + (n0 + m) * 128;         // row base is 512B aligned
    v2f bf[32], af[32];
#pragma unroll
    for (int s = 0; s < 32; ++s) {
      const int ka = 4 * s + 2 * half;
      bf[s] = *(const v2f*)(Brow + ka);              // global_load_b64, deep pipeline
    }
#pragma unroll
    for (int s = 0; s < 32; ++s) {
      const int ka = 4 * s + 2 * half;
      af[s] = *(const v2f*)(h1 + ka) * amask;        // ds_load_b64 + mask
    }
    v8f c = {};
#pragma unroll
    for (int s = 0; s < 32; ++s)
      c = __builtin_amdgcn_wmma_f32_16x16x4_f32(false, af[s], false, bf[s], (short)0, c, false, false);
    if (half == 0) {
      const int n = n0 + m;
      h2[n] = fmaxf(c[0] + b2[n], 0.f);
    }
  }
  __syncthreads();

  // ---- Layer 3: joint_F = h2 @ W3^T + b3. N=69 (5 waves cover 80), K=128 ----
  // Columns n>=69 only affect C entries we never store: clamp row, guard store.
  if (wave < 5) {
    const int n0 = wave * 16;
    const int n  = n0 + m;
    const float* Brow = W3 + ((n < 69) ? n : 0) * 128;
    v2f bf[32], af[32];
#pragma unroll
    for (int s = 0; s < 32; ++s) {
      const int ka = 4 * s + 2 * half;
      bf[s] = *(const v2f*)(Brow + ka);
    }
#pragma unroll
    for (int s = 0; s < 32; ++s) {
      const int ka = 4 * s + 2 * half;
      af[s] = *(const v2f*)(h2 + ka) * amask;
    }
    v8f c = {};
#pragma unroll
    for (int s = 0; s < 32; ++s)
      c = __builtin_amdgcn_wmma_f32_16x16x4_f32(false, af[s], false, bf[s], (short)0, c, false, false);
    if (half == 0 && n < 69) jF[n] = c[0] + b3[n];
  }
  __syncthreads();

  // ---- Per-joint tail: gather -> W_pose matvec -> Rodrigues -> 3x3 SVD ----
  if (tid < NJ) {
    const int j = tid;
    float xin[21];
#pragma unroll
    for (int p = 0; p < MAXC; ++p) {
      const int idx = c_idx[j][p];
#pragma unroll
      for (int d = 0; d < 3; ++d)
        xin[p * 3 + d] = (idx >= 0) ? jF[idx * 3 + d] : 0.f;
    }

    float rv[3];
#pragma unroll
    for (int d = 0; d < 3; ++d) {
      float acc = bp[j * 3 + d];
      const float* wr = Wp + (j * 3 + d) * 21;
#pragma unroll
      for (int k = 0; k < 21; ++k) acc += wr[k] * xin[k];
      rv[d] = acc;
    }

    const float th  = sqrtf(1e-5f + rv[0]*rv[0] + rv[1]*rv[1] + rv[2]*rv[2]);
    const float inv = 1.f / th;
    const float x = rv[0]*inv, y = rv[1]*inv, z = rv[2]*inv;
    const float ct = cosf(th), st = sinf(th), oc = 1.f - ct;

    float R[3][3];
    R[0][0] = x*x + (1.f - x*x)*ct;  R[0][1] = x*y*oc - z*st;        R[0][2] = x*z*oc + y*st;
    R[1][0] = x*y*oc + z*st;         R[1][1] = y*y + (1.f - y*y)*ct; R[1][2] = y*z*oc - x*st;
    R[2][0] = x*z*oc - y*st;         R[2][1] = y*z*oc + x*st;        R[2][2] = z*z + (1.f - z*z)*ct;

#pragma unroll
    for (int a = 0; a < 3; ++a)
#pragma unroll
      for (int b = 0; b < 3; ++b)
        out[j * 9 + a * 3 + b] = R[a][b];

    // SVD via eigen-decomposition of R^T R (symmetric, near identity).
    float M[3][3];
#pragma unroll
    for (int a = 0; a < 3; ++a)
#pragma unroll
      for (int b = 0; b < 3; ++b)
        M[a][b] = R[0][a]*R[0][b] + R[1][a]*R[1][b] + R[2][a]*R[2][b];

    float V[3][3] = {{1.f,0.f,0.f},{0.f,1.f,0.f},{0.f,0.f,1.f}};
    for (int sweep = 0; sweep < 6; ++sweep) {
      jacobi_rot(M, V, 0, 1);
      jacobi_rot(M, V, 0, 2);
      jacobi_rot(M, V, 1, 2);
    }

    float w[3] = {M[0][0], M[1][1], M[2][2]};
    int o0 = 0, o1 = 1, o2 = 2, tswp;
    if (w[o0] < w[o1]) { tswp = o0; o0 = o1; o1 = tswp; }
    if (w[o0] < w[o2]) { tswp = o0; o0 = o2; o2 = tswp; }
    if (w[o1] < w[o2]) { tswp = o1; o1 = o2; o2 = tswp; }
    const int ord[3] = {o0, o1, o2};

#pragma unroll
    for (int k = 0; k < 3; ++k) {
      const int e = ord[k];
      const float sv = sqrtf(fmaxf(w[e], 0.f));
      out[414 + j * 3 + k] = sv;                       // S
      const float is = (sv > 1e-12f) ? 1.f / sv : 0.f;
#pragma unroll
      for (int a = 0; a < 3; ++a) {
        const float u = (R[a][0]*V[0][e] + R[a][1]*V[1][e] + R[a][2]*V[2][e]) * is;
        out[207 + j * 9 + a * 3 + k] = u;              // U column k
        out[483 + j * 9 + a * 3 + k] = V[a][e];        // V column k
      }
    }
  }
}

extern "C" void kernel_launch(void* const* d_in, const int* in_sizes, int n_in,
                              void* d_out, int out_size, void* d_ws, size_t ws_size,
                              hipStream_t stream) {
  (void)in_sizes; (void)n_in; (void)out_size; (void)d_ws; (void)ws_size;
  const float* feature = (const float*)d_in[0];
  const float* W1 = (const float*)d_in[1];
  const float* b1 = (const float*)d_in[2];
  const float* W2 = (const float*)d_in[3];
  const float* b2 = (const float*)d_in[4];
  const float* W3 = (const float*)d_in[5];
  const float* b3 = (const float*)d_in[6];
  const float* Wp = (const float*)d_in[7];
  const float* bp = (const float*)d_in[8];
  float* out = (float*)d_out;
  pose_fused_kernel<<<1, 256, 0, stream>>>(feature, W1, b1, W2, b2, W3, b3, Wp, bp, out);
}